// MaxPoolingSequenceSpans_1872605741238
// MI455X (gfx1250) — compile-verified
//
#include <hip/hip_runtime.h>

// Problem constants (from the reference):
//   context:     (B=8, S=4096, H=1024) float32
//   spans_begin: (B, N_SPANS=100) int32
//   spans_len:   (B, N_SPANS=100) int32
//   out:         (B, N_SPANS, H)  float32 = elementwise max over span rows
constexpr int kB     = 8;
constexpr int kS     = 4096;
constexpr int kH     = 1024;
constexpr int kSpans = 100;
constexpr int kRowV4 = kH / 4;   // 256 float4 chunks per row
constexpr int kChunk = 8;        // rows staged per async chunk

typedef float v4f __attribute__((ext_vector_type(4)));

// gfx1250 async DMA: global -> LDS, 16 bytes per lane, tracked by ASYNCcnt.
// VDST = per-lane LDS byte address (low 32 bits of the flat LDS pointer),
// VADDR = per-lane 64-bit global address, saddr disabled ("off").
__device__ __forceinline__ void async_row_to_lds(const v4f* gsrc, v4f* ldst) {
    asm volatile("global_load_async_to_lds_b128 %0, %1, off"
                 :: "v"((unsigned)(size_t)ldst), "v"(gsrc)
                 : "memory");
}

__global__ __launch_bounds__(256)
void span_maxpool_kernel(const float* __restrict__ ctx,
                         const int*  __restrict__ spans_begin,
                         const int*  __restrict__ spans_len,
                         float*      __restrict__ out) {
    // Double-buffered 8-row tile: 2 * 8 * 256 lanes * 16 B = 64 KB LDS.
    __shared__ v4f buf[2][kChunk][256];

    const int n  = blockIdx.x;            // span index
    const int b  = blockIdx.y;            // batch index
    const int sp = b * kSpans + n;        // flat (b, span)

    // Uniform across the block -> scalar loads.
    const int bg = spans_begin[sp];
    int ln = spans_len[sp];
    ln = (ln < 1) ? 1 : ln;               // eff_len = max(len, 1)

    const v4f* __restrict__ cvec =
        reinterpret_cast<const v4f*>(ctx + (size_t)b * kS * kH);
    const int col = threadIdx.x;          // this lane's float4 column chunk

    v4f m;
    m.x = m.y = m.z = m.w = -__builtin_inff();

    const int nchunks = ln / kChunk;      // full 8-row chunks via async DMA

    auto issue_chunk = [&](int c) {
        // WAR guard: async-DMA LDS writes are NOT ordered with ds_load, so
        // make sure prior reads of the buffer we are about to overwrite have
        // retired. In steady state this is already satisfied (their data fed
        // the max chain an iteration ago).
        asm volatile("s_wait_dscnt 0x0" ::: "memory");
        const int base = bg + c * kChunk;
#pragma unroll
        for (int u = 0; u < kChunk; ++u) {
            int row = base + u;
            row = (row < 0) ? 0 : row;
            row = (row > kS - 1) ? (kS - 1) : row;   // clip (matches reference)
            async_row_to_lds(&cvec[(size_t)row * kRowV4 + col],
                             &buf[c & 1][u][col]);
        }
    };

    if (nchunks > 0) issue_chunk(0);
    for (int c = 0; c < nchunks; ++c) {
        if (c + 1 < nchunks) {
            issue_chunk(c + 1);                       // prefetch next tile
            // 16 async ops outstanding; in-order completion means waiting
            // down to 8 guarantees chunk c has fully landed in LDS.
            asm volatile("s_wait_asynccnt 0x8" ::: "memory");
        } else {
            asm volatile("s_wait_asynccnt 0x0" ::: "memory");
        }
#pragma unroll
        for (int u = 0; u < kChunk; ++u) {
            const v4f v = buf[c & 1][u][col];         // ds_load_b128, own slot
            m.x = fmaxf(m.x, v.x);
            m.y = fmaxf(m.y, v.y);
            m.z = fmaxf(m.z, v.z);
            m.w = fmaxf(m.w, v.w);
        }
    }

    // Remainder rows (len % 8) via direct b128 loads.
    for (int r = nchunks * kChunk; r < ln; ++r) {
        int row = bg + r;
        row = (row < 0) ? 0 : row;
        row = (row > kS - 1) ? (kS - 1) : row;
        const v4f v = cvec[(size_t)row * kRowV4 + col];
        m.x = fmaxf(m.x, v.x);
        m.y = fmaxf(m.y, v.y);
        m.z = fmaxf(m.z, v.z);
        m.w = fmaxf(m.w, v.w);
    }

    // Write-once output: nontemporal b128 store keeps the 192 MB L2 free for
    // the 128 MB context working set (spans overlap -> L2 re-hits are free).
    v4f* ovec = reinterpret_cast<v4f*>(out);
    __builtin_nontemporal_store(m, &ovec[(size_t)sp * kRowV4 + col]);
}

extern "C" void kernel_launch(void* const* d_in, const int* in_sizes, int n_in,
                              void* d_out, int out_size, void* d_ws, size_t ws_size,
                              hipStream_t stream) {
    const float* ctx = (const float*)d_in[0];
    const int*   sb  = (const int*)d_in[1];
    const int*   sl  = (const int*)d_in[2];
    float*       out = (float*)d_out;

    const dim3 grid(kSpans, kB);    // 800 workgroups, one per (b, span)
    const dim3 block(256);          // 8 wave32 waves
    hipLaunchKernelGGL(span_maxpool_kernel, grid, block, 0, stream,
                       ctx, sb, sl, out);
}